// Att_decoder_f_24189255811766
// MI455X (gfx1250) — compile-verified
//
#include <hip/hip_runtime.h>

typedef __attribute__((ext_vector_type(2))) float v2f;
typedef __attribute__((ext_vector_type(8))) float v8f;

#define SEQ 256
#define DIMV 2
#define FEED 10
#define EPSV 1e-5f

// ds_swizzle_b32 BITMASK_PERM patterns: offset[14:10]=xor, [4:0]=and(0x1F).
// xor masks 1,2,4,8 never cross a 16-lane half.
#define SWZ_XOR8 0x201f
#define SWZ_XOR4 0x101f
#define SWZ_XOR2 0x081f
#define SWZ_XOR1 0x041f

// One butterfly step over a batch of 24 values: 24 back-to-back swizzles
// (DS pipe, in-order returns) followed by 24 adds -> graduated waits.
template <int PAT>
__device__ __forceinline__ void bfly24_step(float (&red)[24]) {
  int tmp[24];
#pragma unroll
  for (int i = 0; i < 24; ++i)
    tmp[i] = __builtin_amdgcn_ds_swizzle(__float_as_int(red[i]), PAT);
#pragma unroll
  for (int i = 0; i < 24; ++i) red[i] += __int_as_float(tmp[i]);
}

__device__ __forceinline__ void bfly24_sum(float (&red)[24]) {
  bfly24_step<SWZ_XOR8>(red);
  bfly24_step<SWZ_XOR4>(red);
  bfly24_step<SWZ_XOR2>(red);
  bfly24_step<SWZ_XOR1>(red);
}

// One wave computes attention output rows [tile*16, tile*16+16) for one batch.
// qs is pre-scaled by log2(e)/sqrt(E), so scores come out of the WMMA already
// in log2 domain and softmax uses the native v_exp_f32 (2^x) directly:
//   exp(s)/sum(exp(s)) == exp2(c*s)/sum(exp2(c*s))  (base change, c folded in Q)
// Scores via V_WMMA_F32_16X16X4_F32 (K padded 2->4 by zeroing the upper-half
// lane slots with a mask multiply). No max-subtraction: LN(dim=2) outputs are
// in [-1,1] and weights are small 2x2 matrices, so |score| is O(1) -- fp32
// exp2 is safe and softmax is shift-invariant.
__device__ __forceinline__ void attention_tile(const v2f* qs, const v2f* ks,
                                               const v2f* vs, v2f* os,
                                               int tile, int lane) {
  const int half = lane >> 4;           // 0: K=0..1 slots, 1: K=2..3 (zero pad)
  const int l = lane & 15;
  const float hm = half ? 0.0f : 1.0f;  // lane mask, no control flow

  // A fragment: 16x4 fp32; lanes0-15 carry (K0,K1) of row M=tile*16+l
  v2f a = qs[tile * 16 + l];
  a.x *= hm;
  a.y *= hm;

  // Preload all B fragments (K rows) and V rows with unpredicated b64 LDS loads.
  v2f bf[16];
  v2f vv[16];
#pragma unroll
  for (int j = 0; j < 16; ++j) {
    v2f b = ks[j * 16 + l];
    bf[j].x = b.x * hm;
    bf[j].y = b.y * hm;
    vv[j] = vs[j * 16 + l];
  }

  // 16 back-to-back WMMAs -> full 16x256 score strip (log2 domain) in registers
  v8f c[16];
#pragma unroll
  for (int j = 0; j < 16; ++j) {
    v8f z = {};
    c[j] = __builtin_amdgcn_wmma_f32_16x16x4_f32(false, a, false, bf[j],
                                                 (short)0, z, false, false);
  }

  // C layout: element r of c[j] = S[M = tile*16 + 8*half + r][N = 16*j + l].
  // Fused exp2 -> rowsum & packed P.V partials, then one batched butterfly.
  float red[24];  // [0:8)=expsum, [8:16)=o0, [16:24)=o1 per row r
#pragma unroll
  for (int r = 0; r < 8; ++r) {
    float sm = 0.0f;
    v2f ov = {0.0f, 0.0f};
#pragma unroll
    for (int j = 0; j < 16; ++j) {
      const float e = __builtin_amdgcn_exp2f(c[j][r]);  // native v_exp_f32
      sm += e;
      ov += vv[j] * e;  // v_pk_fma_f32 (packed fp32 FMA, e splatted)
    }
    red[r] = sm;
    red[8 + r] = ov.x;
    red[16 + r] = ov.y;
  }
  bfly24_sum(red);

#pragma unroll
  for (int r = 0; r < 8; ++r) {
    const float inv = __builtin_amdgcn_rcpf(red[r]);  // 1-ulp reciprocal
    if (l == 0) {
      v2f o;
      o.x = red[8 + r] * inv;
      o.y = red[16 + r] * inv;
      os[tile * 16 + 8 * half + r] = o;
    }
  }
}

__global__ __launch_bounds__(256) void att_decoder_kernel(
    const float* __restrict__ x, const float* __restrict__ m,
    const float* __restrict__ sa_w_in, const float* __restrict__ sa_b_in,
    const float* __restrict__ sa_w_out, const float* __restrict__ sa_b_out,
    const float* __restrict__ ca_w_in, const float* __restrict__ ca_b_in,
    const float* __restrict__ ca_w_out, const float* __restrict__ ca_b_out,
    const float* __restrict__ ln1_g, const float* __restrict__ ln1_b,
    const float* __restrict__ ln2_g, const float* __restrict__ ln2_b,
    const float* __restrict__ ln3_g, const float* __restrict__ ln3_b,
    const float* __restrict__ f_w1, const float* __restrict__ f_b1,
    const float* __restrict__ f_ln_g, const float* __restrict__ f_ln_b,
    const float* __restrict__ f_w2, const float* __restrict__ f_b2,
    float* __restrict__ out) {
  __shared__ v2f h[SEQ];
  __shared__ v2f qb[SEQ];
  __shared__ v2f kb[SEQ];
  __shared__ v2f vb[SEQ];
  __shared__ v2f ob[SEQ];

  const int t = threadIdx.x;       // 0..255, one sequence row each
  const int lane = t & 31;
  const int wave = t >> 5;         // 0..7
  const int batch = blockIdx.x;    // 0..1023
  // log2(e) / sqrt(DIM): base-2 softmax + 1/sqrt(E) attention scaling, both
  // folded into the Q projection (scores are linear in Q).
  const float scale = 1.4426950408889634f * 0.70710678118654752f;

  const v2f* xb = (const v2f*)(x + (size_t)batch * SEQ * DIMV);
  const v2f* mb = (const v2f*)(m + (size_t)batch * SEQ * DIMV);
  v2f* outb = (v2f*)(out + (size_t)batch * SEQ * DIMV);

  // ---- LN1 (over DIM=2) ----
  {
    const v2f xr = xb[t];
    const float mu = 0.5f * (xr.x + xr.y);
    const float d0 = xr.x - mu, d1 = xr.y - mu;
    const float rs = rsqrtf(0.5f * (d0 * d0 + d1 * d1) + EPSV);
    v2f hr;
    hr.x = d0 * rs * ln1_g[0] + ln1_b[0];
    hr.y = d1 * rs * ln1_g[1] + ln1_b[1];
    h[t] = hr;
  }
  __syncthreads();

  // ---- self attention: q,k,v projections (w_in rows: q=0..1, k=2..3, v=4..5) ----
  {
    const v2f hr = h[t];
    v2f q, k, v;
    q.x = (hr.x * sa_w_in[0] + hr.y * sa_w_in[1] + sa_b_in[0]) * scale;
    q.y = (hr.x * sa_w_in[2] + hr.y * sa_w_in[3] + sa_b_in[1]) * scale;
    k.x = hr.x * sa_w_in[4] + hr.y * sa_w_in[5] + sa_b_in[2];
    k.y = hr.x * sa_w_in[6] + hr.y * sa_w_in[7] + sa_b_in[3];
    v.x = hr.x * sa_w_in[8] + hr.y * sa_w_in[9] + sa_b_in[4];
    v.y = hr.x * sa_w_in[10] + hr.y * sa_w_in[11] + sa_b_in[5];
    qb[t] = q;
    kb[t] = k;
    vb[t] = v;
  }
  __syncthreads();
  attention_tile(qb, kb, vb, ob, wave, lane);
  attention_tile(qb, kb, vb, ob, wave + 8, lane);
  __syncthreads();

  // ---- out-proj + residual + LN2 ----
  {
    const v2f o = ob[t];
    const v2f hr = h[t];
    const float a0 = hr.x + o.x * sa_w_out[0] + o.y * sa_w_out[1] + sa_b_out[0];
    const float a1 = hr.y + o.x * sa_w_out[2] + o.y * sa_w_out[3] + sa_b_out[1];
    const float mu = 0.5f * (a0 + a1);
    const float d0 = a0 - mu, d1 = a1 - mu;
    const float rs = rsqrtf(0.5f * (d0 * d0 + d1 * d1) + EPSV);
    __syncthreads(); // all reads of old h done before overwrite
    v2f hn;
    hn.x = d0 * rs * ln2_g[0] + ln2_b[0];
    hn.y = d1 * rs * ln2_g[1] + ln2_b[1];
    h[t] = hn;
  }
  __syncthreads();

  // ---- cross attention: q from h, k/v from m ----
  {
    const v2f hr = h[t];
    v2f q;
    q.x = (hr.x * ca_w_in[0] + hr.y * ca_w_in[1] + ca_b_in[0]) * scale;
    q.y = (hr.x * ca_w_in[2] + hr.y * ca_w_in[3] + ca_b_in[1]) * scale;
    qb[t] = q;
    const v2f mr = mb[t];
    v2f k, v;
    k.x = mr.x * ca_w_in[4] + mr.y * ca_w_in[5] + ca_b_in[2];
    k.y = mr.x * ca_w_in[6] + mr.y * ca_w_in[7] + ca_b_in[3];
    v.x = mr.x * ca_w_in[8] + mr.y * ca_w_in[9] + ca_b_in[4];
    v.y = mr.x * ca_w_in[10] + mr.y * ca_w_in[11] + ca_b_in[5];
    kb[t] = k;
    vb[t] = v;
  }
  __syncthreads();
  attention_tile(qb, kb, vb, ob, wave, lane);
  attention_tile(qb, kb, vb, ob, wave + 8, lane);
  __syncthreads();

  // ---- out-proj + residual + LN3 (h3 stays in registers) ----
  float h30, h31;
  {
    const v2f o = ob[t];
    const v2f hr = h[t];
    const float a0 = hr.x + o.x * ca_w_out[0] + o.y * ca_w_out[1] + ca_b_out[0];
    const float a1 = hr.y + o.x * ca_w_out[2] + o.y * ca_w_out[3] + ca_b_out[1];
    const float mu = 0.5f * (a0 + a1);
    const float d0 = a0 - mu, d1 = a1 - mu;
    const float rs = rsqrtf(0.5f * (d0 * d0 + d1 * d1) + EPSV);
    h30 = d0 * rs * ln3_g[0] + ln3_b[0];
    h31 = d1 * rs * ln3_g[1] + ln3_b[1];
  }

  // ---- FFN: Linear(2->10) -> LN(10) -> ReLU -> Linear(10->2), residual ----
  {
    float ff[FEED];
    float s = 0.0f;
#pragma unroll
    for (int e = 0; e < FEED; ++e) {
      ff[e] = h30 * f_w1[e * 2 + 0] + h31 * f_w1[e * 2 + 1] + f_b1[e];
      s += ff[e];
    }
    const float mu = s * (1.0f / FEED);
    float vs = 0.0f;
#pragma unroll
    for (int e = 0; e < FEED; ++e) {
      const float d = ff[e] - mu;
      vs += d * d;
      ff[e] = d;
    }
    const float rs = rsqrtf(vs * (1.0f / FEED) + EPSV);
    float r0 = f_b2[0], r1 = f_b2[1];
#pragma unroll
    for (int e = 0; e < FEED; ++e) {
      const float y = fmaxf(ff[e] * rs * f_ln_g[e] + f_ln_b[e], 0.0f);
      r0 = fmaf(y, f_w2[0 * FEED + e], r0);
      r1 = fmaf(y, f_w2[1 * FEED + e], r1);
    }
    v2f o;
    o.x = h30 + r0;
    o.y = h31 + r1;
    outb[t] = o;
  }
}

extern "C" void kernel_launch(void* const* d_in, const int* in_sizes, int n_in,
                              void* d_out, int out_size, void* d_ws, size_t ws_size,
                              hipStream_t stream) {
  (void)in_sizes; (void)n_in; (void)out_size; (void)d_ws; (void)ws_size;
  const float* x        = (const float*)d_in[0];
  const float* m        = (const float*)d_in[1];
  const float* sa_w_in  = (const float*)d_in[2];
  const float* sa_b_in  = (const float*)d_in[3];
  const float* sa_w_out = (const float*)d_in[4];
  const float* sa_b_out = (const float*)d_in[5];
  const float* ca_w_in  = (const float*)d_in[6];
  const float* ca_b_in  = (const float*)d_in[7];
  const float* ca_w_out = (const float*)d_in[8];
  const float* ca_b_out = (const float*)d_in[9];
  const float* ln1_g    = (const float*)d_in[10];
  const float* ln1_b    = (const float*)d_in[11];
  const float* ln2_g    = (const float*)d_in[12];
  const float* ln2_b    = (const float*)d_in[13];
  const float* ln3_g    = (const float*)d_in[14];
  const float* ln3_b    = (const float*)d_in[15];
  const float* f_w1     = (const float*)d_in[16];
  const float* f_b1     = (const float*)d_in[17];
  const float* f_ln_g   = (const float*)d_in[18];
  const float* f_ln_b   = (const float*)d_in[19];
  const float* f_w2     = (const float*)d_in[20];
  const float* f_b2     = (const float*)d_in[21];

  att_decoder_kernel<<<1024, 256, 0, stream>>>(
      x, m, sa_w_in, sa_b_in, sa_w_out, sa_b_out,
      ca_w_in, ca_b_in, ca_w_out, ca_b_out,
      ln1_g, ln1_b, ln2_g, ln2_b, ln3_g, ln3_b,
      f_w1, f_b1, f_ln_g, f_ln_b, f_w2, f_b2,
      (float*)d_out);
}